// ScaledDotProductAttention_38525856645257
// MI455X (gfx1250) — compile-verified
//
#include <hip/hip_runtime.h>
#include <hip/hip_bf16.h>
#include <stdint.h>

typedef __attribute__((ext_vector_type(2))) float v2f;
typedef __attribute__((ext_vector_type(8))) float v8f;

#define LQ 2048
#define LK 2048
#define DH 64
#define MT 32
#define MASK_FILL -1.0e10f

// One workgroup = one (batch, 32-query-row) tile.
// Phase 1: raw scores q@k^T -> 256KB LDS tile (fp32 WMMA 16x16x4),
//          while prefetching the explain_weight/mask streams into L2.
// Phase 2: scale + mask-fill + softmax + explain-weight in LDS; single
//          global write of the attn tile.
// Phase 3: out = p @ v from the LDS tile (fp32 WMMA 16x16x4).
__global__ __launch_bounds__(256, 1)
void sdpa_kernel(const float* __restrict__ qg, const float* __restrict__ kg,
                 const float* __restrict__ vg, const float* __restrict__ ewg,
                 const uint8_t* __restrict__ mg,
                 float* __restrict__ outg, float* __restrict__ attng)
{
    __shared__ __align__(16) float S[MT * LK];   // 256 KB score tile
    __shared__ float red[512];                   // row max/sum scratch

    const int tile = blockIdx.x;
    const int b    = tile / (LQ / MT);
    const int q0   = (tile % (LQ / MT)) * MT;

    const int tid  = threadIdx.x;
    const int lane = tid & 31;     // wave32
    const int wv   = tid >> 5;     // 8 waves
    const int rh   = wv & 1;       // row half (rows rh*16 .. rh*16+15)
    const int cg   = wv >> 1;      // column group (0..3)
    const int lr   = lane & 15;    // row/col within 16
    const int lk2  = lane >> 4;    // selects K pair {0,1} vs {2,3}

    const float* qb = qg + ((size_t)b * LQ + q0) * DH;
    const float* kb = kg + (size_t)b * LK * DH;
    const float* vb = vg + (size_t)b * LK * DH;
    const float*   ewt = ewg + ((size_t)b * LQ + q0) * LK;  // 256 KB stream
    const uint8_t* mts = mg  + ((size_t)b * LQ + q0) * LK;  // 64 KB stream

    // ---- q-tile A fragments: 16 x v2f spanning D=64 in K-steps of 4 ----
    // 16x4 f32 A layout: lane<16 row=lr K={k0,k0+1}; lane>=16 row=lr K={k0+2,k0+3}
    v2f afrag[16];
    {
        const float* qrow = qb + (size_t)(rh * 16 + lr) * DH + 2 * lk2;
        #pragma unroll
        for (int t = 0; t < 16; ++t)
            afrag[t] = *(const v2f*)(qrow + 4 * t);
    }

    // ---- Phase 1: raw scores into LDS (+ L2 prefetch of phase-2 streams) ----
    int it = 0;
    for (int nt = cg; nt < LK / 16; nt += 4, ++it) {
        // Warm L2 with the once-read explain_weight / mask tiles while the
        // WMMA pipeline is busy: 1 cacheline (128B) per thread per iteration.
        if (it < 8) {
            __builtin_prefetch(ewt + ((size_t)it * 256 + tid) * 32, 0, 1);
        } else if (it < 10) {
            __builtin_prefetch(mts + ((size_t)(it - 8) * 256 + tid) * 128, 0, 1);
        }

        const int n0 = nt * 16;
        // B (k^T) fragment is the dual layout: contiguous float2 from row n0+lr
        const float* krow = kb + (size_t)(n0 + lr) * DH + 2 * lk2;
        v8f c = {};
        #pragma unroll
        for (int t = 0; t < 16; ++t) {
            v2f bfrag = *(const v2f*)(krow + 4 * t);
            c = __builtin_amdgcn_wmma_f32_16x16x4_f32(false, afrag[t], false, bfrag,
                                                      (short)0, c, false, false);
        }
        // C layout: lane holds col n=lr, rows j + 8*lk2 (j=0..7)
        float* sd = S + (size_t)(rh * 16 + 8 * lk2) * LK + n0 + lr;
        #pragma unroll
        for (int j = 0; j < 8; ++j)
            sd[(size_t)j * LK] = c[j];
    }
    __syncthreads();

    // ---- Phase 2: scale + mask + softmax + explain reweight ----
    const int row = tid >> 3;      // 0..31
    const int sub = tid & 7;       // 8 threads per row
    float* srow = S + (size_t)row * LK;
    const size_t grow = (size_t)b * LQ + q0 + row;
    const uint8_t* mrow = mg  + grow * LK;
    const float*   erow = ewg + grow * LK;
    float*         arow = attng + grow * LK;

    // sweep 1: scale + mask fill, local max (interleaved float4 chunks)
    float lmax = -3.0e38f;
    for (int i = 0; i < LK / 32; ++i) {
        const int kk = (i * 8 + sub) * 4;
        float4 sv = *(float4*)(srow + kk);
        const unsigned mm = *(const unsigned*)(mrow + kk);
        sv.x = (mm & 0x000000FFu) ? MASK_FILL : sv.x * 0.125f;
        sv.y = (mm & 0x0000FF00u) ? MASK_FILL : sv.y * 0.125f;
        sv.z = (mm & 0x00FF0000u) ? MASK_FILL : sv.z * 0.125f;
        sv.w = (mm & 0xFF000000u) ? MASK_FILL : sv.w * 0.125f;
        *(float4*)(srow + kk) = sv;
        lmax = fmaxf(lmax, fmaxf(fmaxf(sv.x, sv.y), fmaxf(sv.z, sv.w)));
    }
    red[tid] = lmax;
    __syncthreads();
    float rmax = red[row * 8];
    #pragma unroll
    for (int j = 1; j < 8; ++j) rmax = fmaxf(rmax, red[row * 8 + j]);

    // sweep 2: exp + local sum
    float lsum = 0.f;
    for (int i = 0; i < LK / 32; ++i) {
        const int kk = (i * 8 + sub) * 4;
        float4 e = *(float4*)(srow + kk);
        e.x = __expf(e.x - rmax);
        e.y = __expf(e.y - rmax);
        e.z = __expf(e.z - rmax);
        e.w = __expf(e.w - rmax);
        *(float4*)(srow + kk) = e;
        lsum += e.x + e.y + e.z + e.w;
    }
    red[256 + tid] = lsum;
    __syncthreads();
    float rsum = 0.f;
    #pragma unroll
    for (int j = 0; j < 8; ++j) rsum += red[256 + row * 8 + j];
    const float rinv = 1.0f / rsum;

    // sweep 3: normalize * explain_weight; write attn tile once to global
    for (int i = 0; i < LK / 32; ++i) {
        const int kk = (i * 8 + sub) * 4;
        float4 e = *(float4*)(srow + kk);
        const float4 w = *(const float4*)(erow + kk);
        e.x = e.x * rinv * w.x;
        e.y = e.y * rinv * w.y;
        e.z = e.z * rinv * w.z;
        e.w = e.w * rinv * w.w;
        *(float4*)(srow + kk) = e;
        *(float4*)(arow + kk) = e;
    }
    __syncthreads();

    // ---- Phase 3: out tile = p @ v (per wave: 16 rows x 16 cols, K=2048) ----
    v8f oc = {};
    const int d0 = cg * 16;
    const float* vcol = vb + d0 + lr;
    const float* sa = S + (size_t)(rh * 16 + lr) * LK + 2 * lk2;
    #pragma unroll 8
    for (int k0 = 0; k0 < LK; k0 += 4) {
        v2f a = *(const v2f*)(sa + k0);           // ds_load_b64 from score tile
        v2f bf;
        bf[0] = vcol[(size_t)(k0 + 2 * lk2) * DH];
        bf[1] = vcol[(size_t)(k0 + 2 * lk2 + 1) * DH];
        oc = __builtin_amdgcn_wmma_f32_16x16x4_f32(false, a, false, bf,
                                                   (short)0, oc, false, false);
    }
    float* orow = outg + ((size_t)b * LQ + q0 + rh * 16 + 8 * lk2) * DH + d0 + lr;
    #pragma unroll
    for (int j = 0; j < 8; ++j)
        orow[(size_t)j * DH] = oc[j];
}

extern "C" void kernel_launch(void* const* d_in, const int* in_sizes, int n_in,
                              void* d_out, int out_size, void* d_ws, size_t ws_size,
                              hipStream_t stream) {
    const float*   q    = (const float*)d_in[0];
    const float*   k    = (const float*)d_in[1];
    const float*   v    = (const float*)d_in[2];
    const float*   ew   = (const float*)d_in[3];
    const uint8_t* mask = (const uint8_t*)d_in[4];  // numpy bool = 1 byte
    float* out = (float*)d_out;
    const int B = in_sizes[0] / (LQ * DH);
    float* attn = out + (size_t)B * LQ * DH;        // outputs concatenated: (output, attn)
    sdpa_kernel<<<dim3(B * (LQ / MT)), dim3(256), 0, stream>>>(q, k, v, ew, mask, out, attn);
}